// MetaQDA_25735444037963
// MI455X (gfx1250) — compile-verified
//
#include <hip/hip_runtime.h>
#include <hip/hip_bf16.h>

// ---------------------------------------------------------------------------
// MetaQDA on MI455X (gfx1250), fp32 WMMA path + CDNA5 async/TDM data movement.
//
// Pipeline (all on `stream`):
//   K0 stats  : Nc, mu, scale, common                        (VALU, tiny)
//   K1 buildL : Lfac = diag(|triu_diag|) + strict_lower       (elementwise)
//   K2 prior  : prior = Lfac*Lfac^T + kappa*m^T m             (WMMA f32 16x16x4)
//   K3 sigma  : sigma_c = scale_c*(prior + Xc^T Xc - (k+Nc) mu mu^T)
//               Xc staged in LDS via TDM tensor_load_to_lds   (WMMA + TENSORcnt)
//   K4 chol   : sigma_c -> lower Cholesky in place, logdet_c  (VALU)
//   K5 trinv  : L_c -> W_c = L_c^{-1} in place (+ v_c = W_c mu_c)  (VALU)
//   K6 mahal  : quad[q][c] = || query_q * W_c^T - v_c ||^2
//               W chunks double-buffered in LDS via
//               global_load_async_to_lds_b128 (ASYNCcnt), DMA overlapped with
//               the 16-WMMA compute; dominant 154 GFLOP GEMM.
//   K7 score  : out = bias_c - 0.5*(common+D)*log1p(dist/common)
// ---------------------------------------------------------------------------

#define DD   768
#define CC   64
#define NQQ  2048
#define SHOT 16
#define NSUP (CC * SHOT)
#define REGP 0.5f

typedef float v2f __attribute__((ext_vector_type(2)));
typedef float v8f __attribute__((ext_vector_type(8)));
typedef unsigned int v4u __attribute__((ext_vector_type(4)));
typedef int v4i __attribute__((ext_vector_type(4)));
typedef int v8i __attribute__((ext_vector_type(8)));

__device__ __forceinline__ v8f wmma_f32(v2f a, v2f b, v8f c) {
  // V_WMMA_F32_16X16X4_F32: D = A(16x4) * B(4x16) + C(16x16)
  return __builtin_amdgcn_wmma_f32_16x16x4_f32(
      /*neg_a=*/false, a, /*neg_b=*/false, b,
      /*c_mod=*/(short)0, c, /*reuse_a=*/false, /*reuse_b=*/false);
}

// Flat pointer -> wave-relative LDS byte offset (LDS aperture keeps the
// offset in addr[31:0]).
__device__ __forceinline__ unsigned lds_off_u32(const void* p) {
  return (unsigned)(uintptr_t)p;
}

__device__ __forceinline__ void async_ld_b128(unsigned lds_dst,
                                              const void* gsrc) {
  asm volatile("global_load_async_to_lds_b128 %0, %1, off"
               :: "v"(lds_dst), "v"(gsrc) : "memory");
}
__device__ __forceinline__ void wait_async0() {
  asm volatile("s_wait_asynccnt 0x0" ::: "memory");
}

// ---------------------------------------------------------------------------
// K0: per-class stats. One block per class.
// ---------------------------------------------------------------------------
__global__ void stats_kernel(const float* __restrict__ sup,
                             const float* __restrict__ m,
                             const float* __restrict__ kappa_p,
                             const float* __restrict__ nu_p,
                             const int*   __restrict__ labels,
                             float* __restrict__ mu,
                             float* __restrict__ Ncv,
                             float* __restrict__ scale,
                             float* __restrict__ common) {
  const int c   = blockIdx.x;
  const int tid = threadIdx.x;

  __shared__ int slab[NSUP];
  __shared__ int scnt;
  if (tid == 0) scnt = 0;
  for (int n = tid; n < NSUP; n += 256) slab[n] = labels[n];
  __syncthreads();

  int lc = 0;
  for (int n = tid; n < NSUP; n += 256) lc += (slab[n] == c) ? 1 : 0;
  atomicAdd(&scnt, lc);
  __syncthreads();

  const float Nc     = (float)scnt;
  const float kappa_ = fabsf(kappa_p[0]) + 1e-6f;
  const float nu_    = fmaxf(nu_p[0], (float)DD - 1.0f + 1e-6f);
  const float wk     = kappa_ / (kappa_ + Nc);
  const float wn     = Nc / (kappa_ + Nc);
  const float invNc  = (Nc > 0.0f) ? 1.0f / Nc : 0.0f;

  for (int d = tid; d < DD; d += 256) {
    float s = 0.0f;
    for (int n = 0; n < NSUP; n++)
      if (slab[n] == c) s += sup[n * DD + d];
    float mean = s * invNc;
    mu[c * DD + d] = wk * m[d] + wn * mean;
  }

  if (tid == 0) {
    Ncv[c]    = Nc;
    scale[c]  = (kappa_ + Nc + 1.0f) /
                ((nu_ + Nc - (float)DD + 1.0f) * (kappa_ + Nc));
    common[c] = nu_ + Nc + 1.0f - (float)DD;
  }
}

// ---------------------------------------------------------------------------
// K1: materialize the masked prior Cholesky factor.
// ---------------------------------------------------------------------------
__global__ void buildL_kernel(const float* __restrict__ tdiag,
                              const float* __restrict__ tlow,
                              float* __restrict__ Lfac) {
  const int idx = blockIdx.x * 256 + threadIdx.x;
  if (idx >= DD * DD) return;
  const int i = idx / DD;
  const int j = idx % DD;
  float val;
  if (i == j)      val = fabsf(tdiag[i]);
  else if (i > j)  val = tlow[idx];
  else             val = 0.0f;
  Lfac[idx] = val;
}

// ---------------------------------------------------------------------------
// K2: prior = Lfac @ Lfac^T + kappa*m^T m.  One 16x16 tile per wave.
// ---------------------------------------------------------------------------
__global__ void prior_kernel(const float* __restrict__ Lfac,
                             const float* __restrict__ m,
                             const float* __restrict__ kappa_p,
                             float* __restrict__ prior) {
  const int wv   = threadIdx.x >> 5;
  const int lane = threadIdx.x & 31;
  const int half = lane >> 4;
  const int l15  = lane & 15;
  const int t    = blockIdx.x * 8 + wv;       // tile id in [0, 48*48)
  const int tm   = (t / 48) * 16;
  const int tn   = (t % 48) * 16;
  const float* Arow = Lfac + (tm + l15) * DD + half * 2;
  const float* Brow = Lfac + (tn + l15) * DD + half * 2;  // B[k][n] = L[n][k]

  v8f acc = {};
#pragma unroll 4
  for (int k0 = 0; k0 < DD; k0 += 4) {
    v2f a = *(const v2f*)(Arow + k0);
    v2f b = *(const v2f*)(Brow + k0);
    acc = wmma_f32(a, b, acc);
  }

  const float kappa_ = fabsf(kappa_p[0]) + 1e-6f;
  const int   ncol   = tn + l15;
  const float mn     = m[ncol];
  const float* af = reinterpret_cast<const float*>(&acc);
  for (int i = 0; i < 8; i++) {
    const int mm = tm + i + half * 8;
    prior[mm * DD + ncol] = af[i] + kappa_ * m[mm] * mn;
  }
}

// ---------------------------------------------------------------------------
// K3: sigma_c = scale_c * (prior + Xc^T Xc - (kappa+Nc) mu mu^T).
//     Grid: dim3(288, CC). The 16x768 class block is DMA'd into LDS by the
//     Tensor Data Mover (one descriptor issued by wave 0), tracked with
//     TENSORcnt. Fallback: cooperative float4 loads.
// ---------------------------------------------------------------------------
__global__ void sigma_kernel(const float* __restrict__ sup,
                             const float* __restrict__ prior,
                             const float* __restrict__ mu,
                             const float* __restrict__ Ncv,
                             const float* __restrict__ scale,
                             const float* __restrict__ kappa_p,
                             float* __restrict__ sigma) {
  const int c    = blockIdx.y;
  const int wv   = threadIdx.x >> 5;
  const int lane = threadIdx.x & 31;
  const int half = lane >> 4;
  const int l15  = lane & 15;
  const int t    = blockIdx.x * 8 + wv;
  const int tm   = (t / 48) * 16;
  const int tn   = (t % 48) * 16;
  const int mrow = tm + l15;
  const int ncol = tn + l15;
  const int r0   = c * SHOT;

  __shared__ __align__(16) float xs[SHOT * DD];   // 48 KB: X_c, [k][d] row-major

#if __has_builtin(__builtin_amdgcn_tensor_load_to_lds)
  if (wv == 0) {
    // TDM descriptor: 2D tile 16 rows x 768 f32, row stride 768, into xs.
    const unsigned long long ga =
        (unsigned long long)(uintptr_t)(sup + (size_t)r0 * DD);
    v4u g0;
    g0.x = 1u;                                   // count=1 (valid), user mode
    g0.y = lds_off_u32(&xs[0]);                  // lds_addr
    g0.z = (unsigned)(ga & 0xffffffffu);         // global_addr[31:0]
    g0.w = (unsigned)((ga >> 32) & 0x01ffffffu)  // global_addr[56:32]
         | (2u << 30);                           // type=2 ("image")
    v8i g1 = {};
    g1[0] = (int)(2u << 16);                     // data_size=2 -> 4 bytes
    g1[1] = (int)(768u << 16);                   // tensor_dim0[15:0] = 768
    g1[2] = (int)(16u << 16);                    // tensor_dim1[15:0] = 16
    g1[3] = (int)(768u << 16);                   // tile_dim0 = 768
    g1[4] = 16;                                  // tile_dim1 = 16
    g1[5] = 768;                                 // tensor_dim0_stride = 768
    v4i gz = {};
#if __clang_major__ >= 23
    v8i gz8 = {};
    __builtin_amdgcn_tensor_load_to_lds(g0, g1, gz, gz, gz8, 0);
#else
    __builtin_amdgcn_tensor_load_to_lds(g0, g1, gz, gz, 0);
#endif
    __builtin_amdgcn_s_wait_tensorcnt((short)0);
  }
#else
  for (int i = threadIdx.x; i < SHOT * DD / 4; i += 256)
    *(float4*)&xs[i * 4] = *(const float4*)&sup[r0 * DD + i * 4];
#endif
  __syncthreads();

  v8f acc = {};
#pragma unroll
  for (int k0 = 0; k0 < SHOT; k0 += 4) {
    const int kb = k0 + half * 2;
    v2f a, b;
    a.x = xs[kb * DD + mrow];        // A[m][k] = X[k][m]
    a.y = xs[(kb + 1) * DD + mrow];
    b.x = xs[kb * DD + ncol];        // B[k][n] = X[k][n]
    b.y = xs[(kb + 1) * DD + ncol];
    acc = wmma_f32(a, b, acc);
  }

  const float kappa_ = fabsf(kappa_p[0]) + 1e-6f;
  const float coef = kappa_ + Ncv[c];
  const float sc   = scale[c];
  const float mun  = mu[c * DD + ncol];
  float* S = sigma + (size_t)c * DD * DD;
  const float* af = reinterpret_cast<const float*>(&acc);
  for (int i = 0; i < 8; i++) {
    const int mm = tm + i + half * 8;
    const float val = af[i] + prior[mm * DD + ncol]
                    - coef * mu[c * DD + mm] * mun;
    S[mm * DD + ncol] = val * sc;
  }
}

// ---------------------------------------------------------------------------
// K4: in-place lower Cholesky of sigma_c + logdet. One block per class.
// ---------------------------------------------------------------------------
__global__ void chol_kernel(float* __restrict__ sigma,
                            float* __restrict__ logdet) {
  const int c   = blockIdx.x;
  const int tid = threadIdx.x;
  float* A = sigma + (size_t)c * DD * DD;

  for (int k = 0; k < DD; k++) {
    if (tid == 0) A[k * DD + k] = sqrtf(A[k * DD + k]);
    __syncthreads();
    const float dinv = 1.0f / A[k * DD + k];
    for (int i = k + 1 + tid; i < DD; i += 256) A[i * DD + k] *= dinv;
    __syncthreads();
    for (int i = k + 1 + tid; i < DD; i += 256) {
      const float lik = A[i * DD + k];
      for (int j = k + 1; j <= i; j++) A[i * DD + j] -= lik * A[j * DD + k];
    }
    __syncthreads();
  }

  __shared__ float red[256];
  float s = 0.0f;
  for (int k = tid; k < DD; k += 256) s += logf(A[k * DD + k]);
  red[tid] = s;
  __syncthreads();
  for (int off = 128; off > 0; off >>= 1) {
    if (tid < off) red[tid] += red[tid + off];
    __syncthreads();
  }
  if (tid == 0) logdet[c] = 2.0f * red[0];
}

// ---------------------------------------------------------------------------
// K5: in-place triangular inverse W = L^{-1} (row sweep; row i of L staged in
//     LDS before overwrite), zero upper triangle, then v_c = W_c * mu_c.
// ---------------------------------------------------------------------------
__global__ void trinv_kernel(float* __restrict__ sigma,
                             const float* __restrict__ mu,
                             float* __restrict__ v) {
  const int c   = blockIdx.x;
  const int tid = threadIdx.x;
  float* A = sigma + (size_t)c * DD * DD;
  __shared__ float row[DD];

  for (int i = 0; i < DD; i++) {
    for (int j = tid; j <= i; j += 256) row[j] = A[i * DD + j];
    __syncthreads();
    const float dinv = 1.0f / row[i];
    for (int j = tid; j < i; j += 256) {
      float s = 0.0f;
      for (int k = j; k < i; k++) s += row[k] * A[k * DD + j];  // A[k][j]=W[k][j]
      A[i * DD + j] = -dinv * s;
    }
    if (tid == 0) A[i * DD + i] = dinv;
    for (int j = i + 1 + tid; j < DD; j += 256) A[i * DD + j] = 0.0f;
    __syncthreads();
  }

  const float* muc = mu + c * DD;
  for (int i = tid; i < DD; i += 256) {
    float s = 0.0f;
    const float* Wr = A + i * DD;
    for (int j = 0; j <= i; j++) s += Wr[j] * muc[j];
    v[c * DD + i] = s;
  }
}

// ---------------------------------------------------------------------------
// K6: quad[q][c] = || query_q @ W_c^T - v_c ||^2.   Dominant GEMM (154 GFLOP).
//     Grid: dim3(16, CC), 256 thr. All 8 waves share class c; W chunks are
//     DOUBLE-BUFFERED in LDS via GLOBAL_LOAD_ASYNC_TO_LDS_B128: iteration i
//     issues the async copy of chunk i+1 into the alternate buffer, computes
//     16 WMMAs on chunk i, then s_wait_asynccnt 0 + barrier — DMA overlapped
//     with matrix-core work. Row sums stay in registers (deterministic).
// ---------------------------------------------------------------------------
__global__ void mahal_kernel(const float* __restrict__ query,
                             const float* __restrict__ W,
                             const float* __restrict__ v,
                             float* __restrict__ quad) {
  const int c    = blockIdx.y;
  const int wv   = threadIdx.x >> 5;
  const int lane = threadIdx.x & 31;
  const int half = lane >> 4;
  const int l15  = lane & 15;
  const int tm   = (blockIdx.x * 8 + wv) * 16;   // query tile base
  const int arow = tm + l15;

  const float* Wc = W + (size_t)c * DD * DD;
  const float* vc = v + c * DD;
  const float* Aq = query + arow * DD + half * 2;

  // LDS: double-buffered 16(n) x 64(k) chunks of W_c, shared by all 8 waves.
  __shared__ __align__(16) float wbuf[2][16 * 64];   // 2 x 4 KB
  const int ldr = threadIdx.x >> 4;        // 0..15 : n row within tile
  const int ldc = (threadIdx.x & 15) * 4;  // 0..60 : k col (float4)
  const unsigned ldst[2] = { lds_off_u32(&wbuf[0][ldr * 64 + ldc]),
                             lds_off_u32(&wbuf[1][ldr * 64 + ldc]) };

  float rs[8];
  for (int i = 0; i < 8; i++) rs[i] = 0.0f;

  for (int tn = 0; tn < DD; tn += 16) {
    const float* Wrow = &Wc[(tn + ldr) * DD + ldc];

    // Prologue: stage chunk 0 into buffer 0.
    async_ld_b128(ldst[0], Wrow);
    wait_async0();
    __syncthreads();

    v8f acc = {};
    int cur = 0;
    for (int kc = 0; kc < DD; kc += 64) {
      // Issue async stage of the NEXT chunk into the alternate buffer; it
      // proceeds while we run the 16 WMMAs below.
      if (kc + 64 < DD) {
        async_ld_b128(ldst[cur ^ 1], Wrow + kc + 64);
        __builtin_prefetch(Aq + kc + 64, 0, 1);  // A-stream prefetch
      }

      const float* Bk = &wbuf[cur][l15 * 64 + half * 2];
#pragma unroll
      for (int kk = 0; kk < 64; kk += 4) {
        v2f a = *(const v2f*)(Aq + kc + kk);   // A[m][k] = query[q][k]
        v2f b = *(const v2f*)(Bk + kk);        // B[k][n] = W[n][k] (from LDS)
        acc = wmma_f32(a, b, acc);
      }

      wait_async0();        // my async write to the alternate buffer is done
      __syncthreads();      // all waves done reading wbuf[cur]
      cur ^= 1;
    }

    const float vn = vc[tn + l15];
    const float* af = reinterpret_cast<const float*>(&acc);
    for (int i = 0; i < 8; i++) {
      float y  = af[i] - vn;
      float sq = y * y;
      sq += __shfl_xor(sq, 1, 32);
      sq += __shfl_xor(sq, 2, 32);
      sq += __shfl_xor(sq, 4, 32);
      sq += __shfl_xor(sq, 8, 32);             // sum over the 16-lane half
      if (l15 == 0) rs[i] += sq;
    }
  }
  if (l15 == 0)
    for (int i = 0; i < 8; i++)
      quad[(size_t)(tm + i + half * 8) * CC + c] = rs[i];
}

// ---------------------------------------------------------------------------
// K7: final scoring.
// ---------------------------------------------------------------------------
__global__ void score_kernel(const float* __restrict__ query,
                             const float* __restrict__ mu,
                             const float* __restrict__ quad,
                             const float* __restrict__ common,
                             const float* __restrict__ logdet,
                             float* __restrict__ out) {
  const int idx = blockIdx.x * blockDim.x + threadIdx.x;
  if (idx >= NQQ * CC) return;
  const int q = idx / CC;
  const int c = idx % CC;

  float s = 0.0f;
  for (int d = 0; d < DD; d++) {
    const float t = query[q * DD + d] - mu[c * DD + d];
    s += t * t;
  }
  const float dist = (1.0f - REGP) * quad[idx] + REGP * s;
  const float com  = common[c];
  const float bias = lgammaf(0.5f * (com + (float)DD)) - lgammaf(0.5f * com)
                   - 0.5f * (float)DD * logf(com) - 0.5f * logdet[c];
  out[idx] = bias - 0.5f * (com + (float)DD) * log1pf(dist / com);
}

// ---------------------------------------------------------------------------
extern "C" void kernel_launch(void* const* d_in, const int* in_sizes, int n_in,
                              void* d_out, int out_size, void* d_ws, size_t ws_size,
                              hipStream_t stream) {
  const float* sup    = (const float*)d_in[0];  // [1024, 768]
  const float* query  = (const float*)d_in[1];  // [2048, 768]
  const float* m      = (const float*)d_in[2];  // [1, 768]
  const float* kappa  = (const float*)d_in[3];  // scalar
  const float* nu     = (const float*)d_in[4];  // scalar
  const float* tdiag  = (const float*)d_in[5];  // [768]
  const float* tlow   = (const float*)d_in[6];  // [768, 768]
  const int*   labels = (const int*)  d_in[7];  // [1024]
  float* out = (float*)d_out;                   // [2048, 64]

  // Workspace layout (floats). Needs ~156.7 MB.
  float* w      = (float*)d_ws;
  float* sigma  = w;                                  // [C][D][D] -> later W
  float* prior  = sigma  + (size_t)CC * DD * DD;      // [D][D]
  float* mu     = prior  + (size_t)DD * DD;           // [C][D]
  float* v      = mu     + (size_t)CC * DD;           // [C][D]
  float* Ncv    = v      + (size_t)CC * DD;           // [C]
  float* scale  = Ncv    + CC;                        // [C]
  float* common = scale  + CC;                        // [C]
  float* logdet = common + CC;                        // [C]
  float* quad   = logdet + CC;                        // [Q][C]
  float* Lfac   = quad   + (size_t)NQQ * CC;          // [D][D]

  stats_kernel<<<CC, 256, 0, stream>>>(sup, m, kappa, nu, labels,
                                       mu, Ncv, scale, common);
  buildL_kernel<<<(DD * DD + 255) / 256, 256, 0, stream>>>(tdiag, tlow, Lfac);
  prior_kernel<<<288, 256, 0, stream>>>(Lfac, m, kappa, prior);
  sigma_kernel<<<dim3(288, CC), 256, 0, stream>>>(sup, prior, mu, Ncv, scale,
                                                  kappa, sigma);
  chol_kernel<<<CC, 256, 0, stream>>>(sigma, logdet);
  trinv_kernel<<<CC, 256, 0, stream>>>(sigma, mu, v);
  mahal_kernel<<<dim3(16, CC), 256, 0, stream>>>(query, sigma, v, quad);
  score_kernel<<<(NQQ * CC + 255) / 256, 256, 0, stream>>>(query, mu, quad,
                                                           common, logdet, out);
}